// TensorMessagePassingNet_51144470561240
// MI455X (gfx1250) — compile-verified
//
#include <hip/hip_runtime.h>

// ---------------------------------------------------------------------------
// PaiNN-style MPNN forward + coordinate-gradient (forces) for MI455X/gfx1250.
//  * Node-level [N,64]x[64,64] GEMMs -> V_WMMA_F32_16X16X4_F32 (wave32 tiles)
//  * Weights staged in LDS pitch-65 so each B fragment is one contiguous
//    ds_load_b64 and across-lane bank index = (n+k)%64 (conflict-free)
//  * Edge-level gather/scatter -> scalar threads + global f32 atomics
//  * Backward derived by hand; only coordinate grads needed (no weight grads)
// ---------------------------------------------------------------------------

#define NCH    64
#define NBASIS 16
#define RCUT   5.0f
#define PI_F   3.14159265358979323846f
#define LDSP   65                    // padded LDS pitch (floats) for weights
#define WSZ    (NCH*LDSP)            // one padded 64x64 matrix in LDS

typedef float v2f __attribute__((ext_vector_type(2)));
typedef float v8f __attribute__((ext_vector_type(8)));

__device__ __forceinline__ float sigm(float x)  { return 1.0f / (1.0f + __expf(-x)); }
__device__ __forceinline__ float dsilu(float x) { float s = sigm(x); return s * (1.0f + x * (1.0f - s)); }
__device__ __forceinline__ v8f vzero8() { v8f z = {0.f,0.f,0.f,0.f,0.f,0.f,0.f,0.f}; return z; }

// D = A(16x4) * B(4x16) + C   (f32 WMMA, wave32). Immediates: no neg, no c-mod.
#define WMMA4(a, b, c) __builtin_amdgcn_wmma_f32_16x16x4_f32(false, (a), false, (b), (short)0, (c), false, false)

// B-fragment fetch from pitch-65 LDS: pair {row k, row k+1} at column n,
// stored so both elements are contiguous (one ds_load_b64).
__device__ __forceinline__ v2f ldsB(const float* sW, int n, int k)
{
    v2f b; b.x = sW[n*LDSP + k]; b.y = sW[n*LDSP + k + 1]; return b;
}

// ---------------------------------------------------------------------------
// Edge geometry: d, rhat, Gaussian RBF, cosine cutoff
// ---------------------------------------------------------------------------
__global__ void k_geom(const float* __restrict__ coord, const int* __restrict__ ei,
                       const int* __restrict__ ej, float* __restrict__ dOut,
                       float* __restrict__ rhatOut, float* __restrict__ rbfOut,
                       float* __restrict__ fcOut, int nE)
{
    int e = blockIdx.x * blockDim.x + threadIdx.x;
    if (e >= nE) return;
    int i = ei[e], j = ej[e];
    float x = coord[3*j+0] - coord[3*i+0];
    float y = coord[3*j+1] - coord[3*i+1];
    float z = coord[3*j+2] - coord[3*i+2];
    float d = sqrtf(x*x + y*y + z*z + 1e-12f);
    float invd = 1.0f / d;
    dOut[e] = d;
    rhatOut[3*e+0] = x*invd; rhatOut[3*e+1] = y*invd; rhatOut[3*e+2] = z*invd;
    const float width = RCUT / NBASIS;          // 0.3125
    const float step  = RCUT / (NBASIS - 1);    // linspace(0,RC,16)
    #pragma unroll
    for (int b = 0; b < NBASIS; ++b) {
        float t = (d - b*step) / width;
        rbfOut[e*NBASIS + b] = __expf(-t*t);
    }
    fcOut[e] = (d < RCUT) ? 0.5f * (__cosf(PI_F * d / RCUT) + 1.0f) : 0.0f;
}

// ---------------------------------------------------------------------------
// Embedding gather: h0_0[n,c] = emb[atomic_number[n], c]
// ---------------------------------------------------------------------------
__global__ void k_embed(const float* __restrict__ emb, const int* __restrict__ an,
                        float* __restrict__ h0, int total)
{
    int idx = blockIdx.x * blockDim.x + threadIdx.x;
    if (idx >= total) return;
    int n = idx >> 6, c = idx & 63;
    h0[idx] = emb[an[n]*NCH + c];
}

// ---------------------------------------------------------------------------
// Forward edge messages: filt = (rbf@Wr + br)*fc ; scatter m0,m1 via atomics
// ---------------------------------------------------------------------------
__global__ void k_edge_fwd(const float* __restrict__ rbf, const float* __restrict__ fcA,
                           const float* __restrict__ rhatA, const int* __restrict__ ei,
                           const int* __restrict__ ej, const float* __restrict__ WrL,
                           const float* __restrict__ brL, const float* __restrict__ h0in,
                           const float* __restrict__ h1in, float* __restrict__ M0,
                           float* __restrict__ M1, int nE)
{
    __shared__ float sWr[4*NBASIS*NCH];   // 16 KB, broadcast reads -> conflict-free
    __shared__ float sbr[4*NCH];
    for (int idx = threadIdx.x; idx < 4*NBASIS*NCH; idx += blockDim.x) sWr[idx] = WrL[idx];
    for (int idx = threadIdx.x; idx < 4*NCH;        idx += blockDim.x) sbr[idx] = brL[idx];
    __syncthreads();

    int e = blockIdx.x * blockDim.x + threadIdx.x;
    if (e >= nE) return;
    float f = fcA[e];
    if (f == 0.0f) return;                 // beyond cutoff: all filters zero
    int i = ei[e], j = ej[e];
    float rb[NBASIS];
    #pragma unroll
    for (int b = 0; b < NBASIS; ++b) rb[b] = rbf[e*NBASIS + b];
    float rx = rhatA[3*e+0], ry = rhatA[3*e+1], rz = rhatA[3*e+2];

    for (int c = 0; c < NCH; ++c) {
        float u0 = sbr[0*NCH+c], u1 = sbr[1*NCH+c], u2 = sbr[2*NCH+c], u3 = sbr[3*NCH+c];
        #pragma unroll
        for (int b = 0; b < NBASIS; ++b) {
            float r = rb[b];
            u0 += r * sWr[(0*NBASIS+b)*NCH + c];
            u1 += r * sWr[(1*NBASIS+b)*NCH + c];
            u2 += r * sWr[(2*NBASIS+b)*NCH + c];
            u3 += r * sWr[(3*NBASIS+b)*NCH + c];
        }
        float f0 = u0*f, f1 = u1*f, f2 = u2*f, f3 = u3*f;
        float h0jc = h0in[(size_t)j*NCH + c];
        float hx = h1in[((size_t)j*NCH + c)*3 + 0];
        float hy = h1in[((size_t)j*NCH + c)*3 + 1];
        float hz = h1in[((size_t)j*NCH + c)*3 + 2];
        float vc = hx*rx + hy*ry + hz*rz;
        atomicAdd(&M0[(size_t)i*NCH + c], f0*h0jc + f2*vc);
        float t1 = f1 * h0jc;
        atomicAdd(&M1[((size_t)i*NCH + c)*3 + 0], t1*rx + f3*hx);
        atomicAdd(&M1[((size_t)i*NCH + c)*3 + 1], t1*ry + f3*hy);
        atomicAdd(&M1[((size_t)i*NCH + c)*3 + 2], t1*rz + f3*hz);
    }
}

// ---------------------------------------------------------------------------
// Weight staging helpers (coalesced global read, conflict-free LDS write)
//   fwd: fetch needs B[k][n]      -> sW[n*65+k] = W[k*64+n]
//   bwd: fetch needs W^T[k][n]    -> sW[n*65+k] = W[n*64+k]
// ---------------------------------------------------------------------------
__device__ __forceinline__ void stageFwd(float* sW, const float* __restrict__ W, int tid, int nthr)
{
    for (int m = tid; m < NCH*NCH; m += nthr) {       // m = k*64+n (row-major read)
        int k = m >> 6, n = m & 63;
        sW[n*LDSP + k] = W[m];
    }
}
__device__ __forceinline__ void stageBwd(float* sW, const float* __restrict__ W, int tid, int nthr)
{
    for (int m = tid; m < NCH*NCH; m += nthr) {       // m = n*64+k (row-major read)
        int n = m >> 6, k = m & 63;
        sW[n*LDSP + k] = W[m];
    }
}

// ---------------------------------------------------------------------------
// Node update, scalar path: h0' = silu(M0@Wmsg0 + h0@Wself0 + b0)   [WMMA]
// ---------------------------------------------------------------------------
__global__ void k_node_fwd0(const float* __restrict__ M0, const float* __restrict__ h0in,
                            const float* __restrict__ Wmsg, const float* __restrict__ Wself,
                            const float* __restrict__ b0, float* __restrict__ a0out,
                            float* __restrict__ h0out, int nAtoms)
{
    __shared__ float sW[2*WSZ];          // 2 x padded 64x64 (33 KB)
    stageFwd(sW,       Wmsg,  threadIdx.x, blockDim.x);
    stageFwd(sW + WSZ, Wself, threadIdx.x, blockDim.x);
    __syncthreads();
    int lane = threadIdx.x & 31, wave = threadIdx.x >> 5;
    int half = lane >> 4, n0 = lane & 15;
    int rowBase = blockIdx.x*64 + wave*16;
    int row = rowBase + n0; if (row >= nAtoms) row = nAtoms - 1;
    const float* A0 = M0   + (size_t)row*NCH;
    const float* A1 = h0in + (size_t)row*NCH;

    v8f acc[4];
    #pragma unroll
    for (int t = 0; t < 4; ++t) acc[t] = vzero8();
    for (int k0 = 0; k0 < NCH; k0 += 4) {
        int k = k0 + 2*half;
        v2f a0v; a0v.x = A0[k]; a0v.y = A0[k+1];
        v2f a1v; a1v.x = A1[k]; a1v.y = A1[k+1];
        #pragma unroll
        for (int t = 0; t < 4; ++t) {
            int n = n0 + 16*t;
            acc[t] = WMMA4(a0v, ldsB(sW,       n, k), acc[t]);
            acc[t] = WMMA4(a1v, ldsB(sW + WSZ, n, k), acc[t]);
        }
    }
    #pragma unroll
    for (int t = 0; t < 4; ++t)
        #pragma unroll
        for (int v = 0; v < 8; ++v) {
            int r = rowBase + v + 8*half;
            int cidx = n0 + 16*t;
            if (r < nAtoms) {
                float a = acc[t][v] + b0[cidx];
                a0out[(size_t)r*NCH + cidx] = a;
                h0out[(size_t)r*NCH + cidx] = a * sigm(a);
            }
        }
}

// ---------------------------------------------------------------------------
// Node update, vector path: h1'[x] = M1[x]@Wmsg1 + h1[x]@Wself1     [WMMA]
// ---------------------------------------------------------------------------
__global__ void k_node_fwd1(const float* __restrict__ M1, const float* __restrict__ h1in,
                            const float* __restrict__ Wmsg, const float* __restrict__ Wself,
                            float* __restrict__ h1out, int nAtoms)
{
    __shared__ float sW[2*WSZ];
    stageFwd(sW,       Wmsg,  threadIdx.x, blockDim.x);
    stageFwd(sW + WSZ, Wself, threadIdx.x, blockDim.x);
    __syncthreads();
    int lane = threadIdx.x & 31, wave = threadIdx.x >> 5;
    int half = lane >> 4, n0 = lane & 15;
    int rowBase = blockIdx.x*64 + wave*16;
    int row = rowBase + n0; if (row >= nAtoms) row = nAtoms - 1;
    const float* A0 = M1   + (size_t)row*NCH*3;
    const float* A1 = h1in + (size_t)row*NCH*3;

    for (int x = 0; x < 3; ++x) {
        v8f acc[4];
        #pragma unroll
        for (int t = 0; t < 4; ++t) acc[t] = vzero8();
        for (int k0 = 0; k0 < NCH; k0 += 4) {
            int k = k0 + 2*half;
            v2f a0v; a0v.x = A0[k*3 + x]; a0v.y = A0[(k+1)*3 + x];
            v2f a1v; a1v.x = A1[k*3 + x]; a1v.y = A1[(k+1)*3 + x];
            #pragma unroll
            for (int t = 0; t < 4; ++t) {
                int n = n0 + 16*t;
                acc[t] = WMMA4(a0v, ldsB(sW,       n, k), acc[t]);
                acc[t] = WMMA4(a1v, ldsB(sW + WSZ, n, k), acc[t]);
            }
        }
        #pragma unroll
        for (int t = 0; t < 4; ++t)
            #pragma unroll
            for (int v = 0; v < 8; ++v) {
                int r = rowBase + v + 8*half;
                int cidx = n0 + 16*t;
                if (r < nAtoms) h1out[((size_t)r*NCH + cidx)*3 + x] = acc[t][v];
            }
    }
}

// ---------------------------------------------------------------------------
// Readout forward: z = h0f@Wro1 + bro1  (store pre-activation for backward)
// ---------------------------------------------------------------------------
__global__ void k_read_fwd(const float* __restrict__ h0, const float* __restrict__ Wro1,
                           const float* __restrict__ bro1, float* __restrict__ z, int nAtoms)
{
    __shared__ float sW[WSZ];
    stageFwd(sW, Wro1, threadIdx.x, blockDim.x);
    __syncthreads();
    int lane = threadIdx.x & 31, wave = threadIdx.x >> 5;
    int half = lane >> 4, n0 = lane & 15;
    int rowBase = blockIdx.x*64 + wave*16;
    int row = rowBase + n0; if (row >= nAtoms) row = nAtoms - 1;
    const float* A = h0 + (size_t)row*NCH;
    v8f acc[4];
    #pragma unroll
    for (int t = 0; t < 4; ++t) acc[t] = vzero8();
    for (int k0 = 0; k0 < NCH; k0 += 4) {
        int k = k0 + 2*half;
        v2f a; a.x = A[k]; a.y = A[k+1];
        #pragma unroll
        for (int t = 0; t < 4; ++t)
            acc[t] = WMMA4(a, ldsB(sW, n0 + 16*t, k), acc[t]);
    }
    #pragma unroll
    for (int t = 0; t < 4; ++t)
        #pragma unroll
        for (int v = 0; v < 8; ++v) {
            int r = rowBase + v + 8*half;
            int cidx = n0 + 16*t;
            if (r < nAtoms) z[(size_t)r*NCH + cidx] = acc[t][v] + bro1[cidx];
        }
}

// ---------------------------------------------------------------------------
// Readout backward: dh0f = (Wro2 * silu'(z)) @ Wro1^T                [WMMA]
// ---------------------------------------------------------------------------
__global__ void k_read_bwd(const float* __restrict__ z, const float* __restrict__ Wro1,
                           const float* __restrict__ Wro2, float* __restrict__ dh0, int nAtoms)
{
    __shared__ float sW[WSZ + NCH];
    stageBwd(sW, Wro1, threadIdx.x, blockDim.x);   // fetch W^T[k][n] = W[n][k]
    for (int idx = threadIdx.x; idx < NCH; idx += blockDim.x) sW[WSZ + idx] = Wro2[idx];
    __syncthreads();
    int lane = threadIdx.x & 31, wave = threadIdx.x >> 5;
    int half = lane >> 4, n0 = lane & 15;
    int rowBase = blockIdx.x*64 + wave*16;
    int row = rowBase + n0; if (row >= nAtoms) row = nAtoms - 1;
    const float* Z = z + (size_t)row*NCH;
    v8f acc[4];
    #pragma unroll
    for (int t = 0; t < 4; ++t) acc[t] = vzero8();
    for (int k0 = 0; k0 < NCH; k0 += 4) {
        int k = k0 + 2*half;
        v2f a; a.x = sW[WSZ + k]   * dsilu(Z[k]);
               a.y = sW[WSZ + k+1] * dsilu(Z[k+1]);
        #pragma unroll
        for (int t = 0; t < 4; ++t)
            acc[t] = WMMA4(a, ldsB(sW, n0 + 16*t, k), acc[t]);
    }
    #pragma unroll
    for (int t = 0; t < 4; ++t)
        #pragma unroll
        for (int v = 0; v < 8; ++v) {
            int r = rowBase + v + 8*half;
            int cidx = n0 + 16*t;
            if (r < nAtoms) dh0[(size_t)r*NCH + cidx] = acc[t][v];
        }
}

// ---------------------------------------------------------------------------
// Node backward, scalar: da0 = dh0out*silu'(a0);
//   dM0 = da0@Wmsg0^T ; dh0in = da0@Wself0^T                          [WMMA]
// ---------------------------------------------------------------------------
__global__ void k_node_bwd0(const float* __restrict__ dh0out, const float* __restrict__ a0,
                            const float* __restrict__ Wmsg, const float* __restrict__ Wself,
                            float* __restrict__ dM0, float* __restrict__ dh0in, int nAtoms)
{
    __shared__ float sW[2*WSZ];
    stageBwd(sW,       Wmsg,  threadIdx.x, blockDim.x);
    stageBwd(sW + WSZ, Wself, threadIdx.x, blockDim.x);
    __syncthreads();
    int lane = threadIdx.x & 31, wave = threadIdx.x >> 5;
    int half = lane >> 4, n0 = lane & 15;
    int rowBase = blockIdx.x*64 + wave*16;
    int row = rowBase + n0; if (row >= nAtoms) row = nAtoms - 1;
    const float* G = dh0out + (size_t)row*NCH;
    const float* A = a0     + (size_t)row*NCH;
    v8f accM[4], accS[4];
    #pragma unroll
    for (int t = 0; t < 4; ++t) { accM[t] = vzero8(); accS[t] = vzero8(); }
    for (int k0 = 0; k0 < NCH; k0 += 4) {
        int k = k0 + 2*half;
        v2f a; a.x = G[k] * dsilu(A[k]); a.y = G[k+1] * dsilu(A[k+1]);
        #pragma unroll
        for (int t = 0; t < 4; ++t) {
            int n = n0 + 16*t;
            accM[t] = WMMA4(a, ldsB(sW,       n, k), accM[t]);
            accS[t] = WMMA4(a, ldsB(sW + WSZ, n, k), accS[t]);
        }
    }
    #pragma unroll
    for (int t = 0; t < 4; ++t)
        #pragma unroll
        for (int v = 0; v < 8; ++v) {
            int r = rowBase + v + 8*half;
            int cidx = n0 + 16*t;
            if (r < nAtoms) {
                dM0  [(size_t)r*NCH + cidx] = accM[t][v];
                dh0in[(size_t)r*NCH + cidx] = accS[t][v];
            }
        }
}

// ---------------------------------------------------------------------------
// Node backward, vector: dM1[x] = dh1out[x]@Wmsg1^T; dh1in[x] = dh1out[x]@Wself1^T
// ---------------------------------------------------------------------------
__global__ void k_node_bwd1(const float* __restrict__ dh1out,
                            const float* __restrict__ Wmsg, const float* __restrict__ Wself,
                            float* __restrict__ dM1, float* __restrict__ dh1in, int nAtoms)
{
    __shared__ float sW[2*WSZ];
    stageBwd(sW,       Wmsg,  threadIdx.x, blockDim.x);
    stageBwd(sW + WSZ, Wself, threadIdx.x, blockDim.x);
    __syncthreads();
    int lane = threadIdx.x & 31, wave = threadIdx.x >> 5;
    int half = lane >> 4, n0 = lane & 15;
    int rowBase = blockIdx.x*64 + wave*16;
    int row = rowBase + n0; if (row >= nAtoms) row = nAtoms - 1;
    const float* G = dh1out + (size_t)row*NCH*3;

    for (int x = 0; x < 3; ++x) {
        v8f accM[4], accS[4];
        #pragma unroll
        for (int t = 0; t < 4; ++t) { accM[t] = vzero8(); accS[t] = vzero8(); }
        for (int k0 = 0; k0 < NCH; k0 += 4) {
            int k = k0 + 2*half;
            v2f a; a.x = G[k*3 + x]; a.y = G[(k+1)*3 + x];
            #pragma unroll
            for (int t = 0; t < 4; ++t) {
                int n = n0 + 16*t;
                accM[t] = WMMA4(a, ldsB(sW,       n, k), accM[t]);
                accS[t] = WMMA4(a, ldsB(sW + WSZ, n, k), accS[t]);
            }
        }
        #pragma unroll
        for (int t = 0; t < 4; ++t)
            #pragma unroll
            for (int v = 0; v < 8; ++v) {
                int r = rowBase + v + 8*half;
                int cidx = n0 + 16*t;
                if (r < nAtoms) {
                    dM1  [((size_t)r*NCH + cidx)*3 + x] = accM[t][v];
                    dh1in[((size_t)r*NCH + cidx)*3 + x] = accS[t][v];
                }
            }
    }
}

// ---------------------------------------------------------------------------
// Backward edge: chain rule through messages, filters, rbf, fc, rhat, d -> forces
// ---------------------------------------------------------------------------
__global__ void k_edge_bwd(const float* __restrict__ rbf, const float* __restrict__ fcA,
                           const float* __restrict__ rhatA, const float* __restrict__ dA,
                           const int* __restrict__ ei, const int* __restrict__ ej,
                           const float* __restrict__ WrL, const float* __restrict__ brL,
                           const float* __restrict__ h0in, const float* __restrict__ h1in,
                           const float* __restrict__ dM0, const float* __restrict__ dM1,
                           float* __restrict__ dh0acc, float* __restrict__ dh1acc,
                           float* __restrict__ forces, int nE)
{
    __shared__ float sWr[4*NBASIS*NCH];
    __shared__ float sbr[4*NCH];
    for (int idx = threadIdx.x; idx < 4*NBASIS*NCH; idx += blockDim.x) sWr[idx] = WrL[idx];
    for (int idx = threadIdx.x; idx < 4*NCH;        idx += blockDim.x) sbr[idx] = brL[idx];
    __syncthreads();

    int e = blockIdx.x * blockDim.x + threadIdx.x;
    if (e >= nE) return;
    float f = fcA[e];
    if (f == 0.0f) return;     // beyond cutoff: every coordinate-grad term vanishes
    int i = ei[e], j = ej[e];
    float d = dA[e];
    float rb[NBASIS], drb[NBASIS];
    #pragma unroll
    for (int b = 0; b < NBASIS; ++b) { rb[b] = rbf[e*NBASIS + b]; drb[b] = 0.0f; }
    float rx = rhatA[3*e+0], ry = rhatA[3*e+1], rz = rhatA[3*e+2];
    float drx = 0.f, dry = 0.f, drz = 0.f, dfc = 0.f;

    for (int c = 0; c < NCH; ++c) {
        float u0 = sbr[0*NCH+c], u1 = sbr[1*NCH+c], u2 = sbr[2*NCH+c], u3 = sbr[3*NCH+c];
        #pragma unroll
        for (int b = 0; b < NBASIS; ++b) {
            float r = rb[b];
            u0 += r * sWr[(0*NBASIS+b)*NCH + c];
            u1 += r * sWr[(1*NBASIS+b)*NCH + c];
            u2 += r * sWr[(2*NBASIS+b)*NCH + c];
            u3 += r * sWr[(3*NBASIS+b)*NCH + c];
        }
        float f0 = u0*f, f1 = u1*f, f2 = u2*f, f3 = u3*f;
        float h0jc = h0in[(size_t)j*NCH + c];
        float hx = h1in[((size_t)j*NCH + c)*3 + 0];
        float hy = h1in[((size_t)j*NCH + c)*3 + 1];
        float hz = h1in[((size_t)j*NCH + c)*3 + 2];
        float vc = hx*rx + hy*ry + hz*rz;
        float g0  = dM0[(size_t)i*NCH + c];
        float g1x = dM1[((size_t)i*NCH + c)*3 + 0];
        float g1y = dM1[((size_t)i*NCH + c)*3 + 1];
        float g1z = dM1[((size_t)i*NCH + c)*3 + 2];
        float s1 = g1x*rx + g1y*ry + g1z*rz;

        float df0 = g0*h0jc;
        float df2 = g0*vc;
        float dv  = g0*f2;
        float df1 = s1*h0jc;
        float df3 = g1x*hx + g1y*hy + g1z*hz;
        float dh0jc = g0*f0 + s1*f1;
        float dh1x = dv*rx + g1x*f3;
        float dh1y = dv*ry + g1y*f3;
        float dh1z = dv*rz + g1z*f3;
        float f1h = f1 * h0jc;
        drx += dv*hx + g1x*f1h;
        dry += dv*hy + g1y*f1h;
        drz += dv*hz + g1z*f1h;
        dfc += df0*u0 + df1*u1 + df2*u2 + df3*u3;
        float du0 = df0*f, du1 = df1*f, du2 = df2*f, du3 = df3*f;
        #pragma unroll
        for (int b = 0; b < NBASIS; ++b) {
            drb[b] += du0 * sWr[(0*NBASIS+b)*NCH + c]
                    + du1 * sWr[(1*NBASIS+b)*NCH + c]
                    + du2 * sWr[(2*NBASIS+b)*NCH + c]
                    + du3 * sWr[(3*NBASIS+b)*NCH + c];
        }
        atomicAdd(&dh0acc[(size_t)j*NCH + c], dh0jc);
        atomicAdd(&dh1acc[((size_t)j*NCH + c)*3 + 0], dh1x);
        atomicAdd(&dh1acc[((size_t)j*NCH + c)*3 + 1], dh1y);
        atomicAdd(&dh1acc[((size_t)j*NCH + c)*3 + 2], dh1z);
    }

    // d-gradient: rbf and cutoff terms (fc>0 => d < RCUT)
    const float width = RCUT / NBASIS;
    const float step  = RCUT / (NBASIS - 1);
    float dd = 0.f;
    #pragma unroll
    for (int b = 0; b < NBASIS; ++b) {
        float t = (d - b*step) / width;
        dd += drb[b] * rb[b] * (-2.0f * t / width);
    }
    dd += dfc * (-0.5f * (PI_F / RCUT) * __sinf(PI_F * d / RCUT));

    // rhat = rij/d ; d = sqrt(|rij|^2+eps) : drij = (drhat - rhat(rhat.drhat))/d + dd*rhat
    float rdot = rx*drx + ry*dry + rz*drz;
    float invd = 1.0f / d;
    float gx = (drx - rx*rdot)*invd + dd*rx;
    float gy = (dry - ry*rdot)*invd + dd*ry;
    float gz = (drz - rz*rdot)*invd + dd*rz;
    // rij = c[j]-c[i]; forces = -dE/dc  =>  F[i]+=g , F[j]-=g
    atomicAdd(&forces[3*i+0],  gx); atomicAdd(&forces[3*i+1],  gy); atomicAdd(&forces[3*i+2],  gz);
    atomicAdd(&forces[3*j+0], -gx); atomicAdd(&forces[3*j+1], -gy); atomicAdd(&forces[3*j+2], -gz);
}

// ---------------------------------------------------------------------------
extern "C" void kernel_launch(void* const* d_in, const int* in_sizes, int n_in,
                              void* d_out, int out_size, void* d_ws, size_t ws_size,
                              hipStream_t stream)
{
    (void)n_in; (void)out_size; (void)ws_size;
    const float* coord  = (const float*)d_in[0];
    const float* emb    = (const float*)d_in[1];
    const float* Wr     = (const float*)d_in[2];   // [L,4,16,64]
    const float* br     = (const float*)d_in[3];   // [L,4,64]
    const float* Wself0 = (const float*)d_in[4];   // [L,64,64]
    const float* Wmsg0  = (const float*)d_in[5];
    const float* b0     = (const float*)d_in[6];   // [L,64]
    const float* Wmsg1  = (const float*)d_in[7];
    const float* Wself1 = (const float*)d_in[8];
    const float* Wro1   = (const float*)d_in[9];
    const float* bro1   = (const float*)d_in[10];
    const float* Wro2   = (const float*)d_in[11];
    const int*   an     = (const int*)d_in[13];
    const int*   ei     = (const int*)d_in[14];
    const int*   ej     = (const int*)d_in[15];
    const int N = in_sizes[0] / 3;
    const int E = in_sizes[14];
    float* F = (float*)d_out;

    // workspace carve-up (floats)
    float* w = (float*)d_ws;
    float* dArr  = w; w += E;
    float* rhat  = w; w += 3*(size_t)E;
    float* rbf   = w; w += NBASIS*(size_t)E;
    float* fc    = w; w += E;
    float* h0_0  = w; w += (size_t)N*NCH;
    float* h0_1  = w; w += (size_t)N*NCH;
    float* h0_2  = w; w += (size_t)N*NCH;
    float* h1_0  = w; w += (size_t)N*NCH*3;
    float* h1_1  = w; w += (size_t)N*NCH*3;
    float* a0_0  = w; w += (size_t)N*NCH;
    float* a0_1  = w; w += (size_t)N*NCH;
    float* zbuf  = w; w += (size_t)N*NCH;
    float* M0    = w; w += (size_t)N*NCH;
    float* M1    = w; w += (size_t)N*NCH*3;
    float* dM0   = w; w += (size_t)N*NCH;
    float* dM1   = w; w += (size_t)N*NCH*3;
    float* dh0a  = w; w += (size_t)N*NCH;
    float* dh0b  = w; w += (size_t)N*NCH;
    float* dh1a  = w; w += (size_t)N*NCH*3;
    float* dh1b  = w; w += (size_t)N*NCH*3;

    dim3 blkE(256), grdE((E + 255) / 256);
    dim3 blkN(128), grdN((N + 63) / 64);
    dim3 blkV(256), grdV(((size_t)N*NCH + 255) / 256);

    hipMemsetAsync(F,    0, (size_t)N*3*sizeof(float), stream);
    hipMemsetAsync(h1_0, 0, (size_t)N*NCH*3*sizeof(float), stream);

    k_geom <<<grdE, blkE, 0, stream>>>(coord, ei, ej, dArr, rhat, rbf, fc, E);
    k_embed<<<grdV, blkV, 0, stream>>>(emb, an, h0_0, N*NCH);

    // ---- layer 0 forward ----
    hipMemsetAsync(M0, 0, (size_t)N*NCH*sizeof(float), stream);
    hipMemsetAsync(M1, 0, (size_t)N*NCH*3*sizeof(float), stream);
    k_edge_fwd <<<grdE, blkE, 0, stream>>>(rbf, fc, rhat, ei, ej, Wr, br, h0_0, h1_0, M0, M1, E);
    k_node_fwd0<<<grdN, blkN, 0, stream>>>(M0, h0_0, Wmsg0, Wself0, b0, a0_0, h0_1, N);
    k_node_fwd1<<<grdN, blkN, 0, stream>>>(M1, h1_0, Wmsg1, Wself1, h1_1, N);

    // ---- layer 1 forward (h1 output of last layer is never consumed: skipped) ----
    hipMemsetAsync(M0, 0, (size_t)N*NCH*sizeof(float), stream);
    hipMemsetAsync(M1, 0, (size_t)N*NCH*3*sizeof(float), stream);
    k_edge_fwd <<<grdE, blkE, 0, stream>>>(rbf, fc, rhat, ei, ej, Wr + 4096, br + 256,
                                           h0_1, h1_1, M0, M1, E);
    k_node_fwd0<<<grdN, blkN, 0, stream>>>(M0, h0_1, Wmsg0 + 4096, Wself0 + 4096, b0 + 64,
                                           a0_1, h0_2, N);

    // ---- readout ----
    k_read_fwd<<<grdN, blkN, 0, stream>>>(h0_2, Wro1, bro1, zbuf, N);

    // ---- backward ----
    hipMemsetAsync(dh1a, 0, (size_t)N*NCH*3*sizeof(float), stream);   // dE/dh1_final = 0
    k_read_bwd<<<grdN, blkN, 0, stream>>>(zbuf, Wro1, Wro2, dh0a, N);

    // layer 1 backward
    k_node_bwd0<<<grdN, blkN, 0, stream>>>(dh0a, a0_1, Wmsg0 + 4096, Wself0 + 4096, dM0, dh0b, N);
    k_node_bwd1<<<grdN, blkN, 0, stream>>>(dh1a, Wmsg1 + 4096, Wself1 + 4096, dM1, dh1b, N);
    k_edge_bwd <<<grdE, blkE, 0, stream>>>(rbf, fc, rhat, dArr, ei, ej, Wr + 4096, br + 256,
                                           h0_1, h1_1, dM0, dM1, dh0b, dh1b, F, E);

    // layer 0 backward
    k_node_bwd0<<<grdN, blkN, 0, stream>>>(dh0b, a0_0, Wmsg0, Wself0, dM0, dh0a, N);
    k_node_bwd1<<<grdN, blkN, 0, stream>>>(dh1b, Wmsg1, Wself1, dM1, dh1a, N);
    k_edge_bwd <<<grdE, blkE, 0, stream>>>(rbf, fc, rhat, dArr, ei, ej, Wr, br,
                                           h0_0, h1_0, dM0, dM1, dh0a, dh1a, F, E);
}